// PerceiverTokenPredictor_12309376271037
// MI455X (gfx1250) — compile-verified
//
#include <hip/hip_runtime.h>
#include <hip/hip_bf16.h>

// Problem shape (from reference): logits [8192, 32000] f32, k = 50.
#define VCOLS    32000
#define NV4      8000            // VCOLS / 4
#define NBINS    2048
#define BIN_SHIFT 21             // 32 - 11 bits -> 2048 bins
#define CAND_CAP 256
#define KEEP_CAP 128
#define CHUNK_V4 256             // 256 lanes * 16B = 4 KB chunk (1024 floats)
#define NFULL    31              // full async chunks per row
#define TAIL_V4  (NV4 - NFULL * CHUNK_V4)   // 64 float4 tail

typedef __attribute__((ext_vector_type(4))) float v4f;
typedef __attribute__((ext_vector_type(4))) int   v4i;
// Pointer types matching the async-LDS builtin's expected parameters:
// arg0: 16-byte vector in global (AS1), arg1: LDS (AS3).
typedef __attribute__((address_space(1))) v4i* gptr_v4i;
typedef __attribute__((address_space(3))) v4i* lptr_v4i;

// Order-preserving float -> uint key (ascending float == ascending uint).
__device__ __forceinline__ unsigned fkey(float f) {
    unsigned u = __float_as_uint(f);
    return (u & 0x80000000u) ? ~u : (u | 0x80000000u);
}

__device__ __forceinline__ float funkey(unsigned k) {
    return __uint_as_float((k & 0x80000000u) ? (k ^ 0x80000000u) : ~k);
}

__global__ __launch_bounds__(256)
void topk_softmax_sample_kernel(const float* __restrict__ logits,
                                const int*   __restrict__ pk,
                                int*         __restrict__ tokens,
                                float*       __restrict__ probs)
{
    __shared__ float    stage[2][CHUNK_V4 * 4];   // async double buffer (8 KB)
    __shared__ unsigned hist[NBINS];              // 8 KB
    __shared__ float    cand_v[CAND_CAP];
    __shared__ int      cand_i[CAND_CAP];
    __shared__ float    abov_v[KEEP_CAP];
    __shared__ int      abov_i[KEEP_CAP];
    __shared__ float    kept_v[KEEP_CAP];
    __shared__ int      kept_i[KEEP_CAP];
    __shared__ float    kept_w[KEEP_CAP];
    __shared__ unsigned s_maxkey, s_ncand, s_nab, s_nkept;
    __shared__ int      s_bbin;
    __shared__ float    s_Z;

    const int tid = threadIdx.x;
    const int row = blockIdx.x;

    int K = *pk;
    K = (K < 1) ? 1 : (K > KEEP_CAP ? KEEP_CAP : K);

    const float* rowp = logits + (size_t)row * VCOLS;
    float*       orow = probs  + (size_t)row * VCOLS;

    for (int b = tid; b < NBINS; b += 256) hist[b] = 0u;
    if (tid == 0) { s_maxkey = 0u; s_ncand = 0u; s_nab = 0u; s_nkept = 0u; }
    __syncthreads();

    unsigned  mymax = 0u;
    const v4f zero4 = (v4f){0.f, 0.f, 0.f, 0.f};

    // ---------------- Pass 1: histogram + row max + zero-fill ----------------
#if __has_builtin(__builtin_amdgcn_global_load_async_to_lds_b128) && \
    __has_builtin(__builtin_amdgcn_s_wait_asynccnt)
    {
        // Double-buffered async global->LDS staging (CDNA5 ASYNCcnt path).
        int buf = 0;
        __builtin_amdgcn_global_load_async_to_lds_b128(
            (gptr_v4i)(rowp + tid * 4),
            (lptr_v4i)(&stage[0][tid * 4]), 0, 0);
        for (int c = 0; c < NFULL; ++c) {
            if (c + 1 < NFULL) {
                __builtin_amdgcn_global_load_async_to_lds_b128(
                    (gptr_v4i)(rowp + (c + 1) * CHUNK_V4 * 4 + tid * 4),
                    (lptr_v4i)(&stage[buf ^ 1][tid * 4]), 0, 0);
                __builtin_amdgcn_s_wait_asynccnt(1);   // chunk c landed
            } else {
                __builtin_amdgcn_s_wait_asynccnt(0);   // last chunk landed
            }
            v4f x = *(const v4f*)(&stage[buf][tid * 4]);
            int base = (c * CHUNK_V4 + tid) * 4;
#pragma unroll
            for (int e = 0; e < 4; ++e) {
                unsigned key = fkey(x[e]);
                mymax = key > mymax ? key : mymax;
                atomicAdd(&hist[key >> BIN_SHIFT], 1u);
            }
            __builtin_nontemporal_store(zero4, (v4f*)(orow + base));
            buf ^= 1;
        }
    }
    if (tid < TAIL_V4) {
        int base = (NFULL * CHUNK_V4 + tid) * 4;
        v4f x = *(const v4f*)(rowp + base);
#pragma unroll
        for (int e = 0; e < 4; ++e) {
            unsigned key = fkey(x[e]);
            mymax = key > mymax ? key : mymax;
            atomicAdd(&hist[key >> BIN_SHIFT], 1u);
        }
        __builtin_nontemporal_store(zero4, (v4f*)(orow + base));
    }
#else
    for (int j = tid; j < NV4; j += 256) {
        int base = j * 4;
        v4f x = *(const v4f*)(rowp + base);
#pragma unroll
        for (int e = 0; e < 4; ++e) {
            unsigned key = fkey(x[e]);
            mymax = key > mymax ? key : mymax;
            atomicAdd(&hist[key >> BIN_SHIFT], 1u);
        }
        __builtin_nontemporal_store(zero4, (v4f*)(orow + base));
    }
#endif
    atomicMax(&s_maxkey, mymax);
    __syncthreads();

    // --------- Find boundary bin: largest bin b with count(>=b) >= K ---------
    if (tid == 0) {
        int b = (int)(s_maxkey >> BIN_SHIFT);
        unsigned above = 0u;
        while (b > 0 && above + hist[b] < (unsigned)K) { above += hist[b]; --b; }
        s_bbin = b;
    }
    __syncthreads();
    const int bbin = s_bbin;

    // ------- Pass 2: gather candidates (last use -> non-temporal re-read) ----
    for (int j = tid; j < NV4; j += 256) {
        v4f x = __builtin_nontemporal_load((const v4f*)(rowp + j * 4));
#pragma unroll
        for (int e = 0; e < 4; ++e) {
            float f = x[e];
            int   b = (int)(fkey(f) >> BIN_SHIFT);
            if (b > bbin) {
                unsigned p = atomicAdd(&s_nab, 1u);
                if (p < KEEP_CAP) { abov_v[p] = f; abov_i[p] = j * 4 + e; }
            } else if (b == bbin) {
                unsigned p = atomicAdd(&s_ncand, 1u);
                if (p < CAND_CAP) { cand_v[p] = f; cand_i[p] = j * 4 + e; }
            }
        }
    }
    __syncthreads();

    int nab = (int)s_nab;   nab = nab > KEEP_CAP ? KEEP_CAP : nab;
    int nc  = (int)s_ncand; nc  = nc  > CAND_CAP ? CAND_CAP : nc;
    int need = K - nab;
    need = need < 0 ? 0 : (need > nc ? nc : need);

    // Everything strictly above the boundary bin is kept.
    for (int j = tid; j < nab; j += 256) {
        unsigned p = atomicAdd(&s_nkept, 1u);
        if (p < KEEP_CAP) { kept_v[p] = abov_v[j]; kept_i[p] = abov_i[j]; }
    }
    // Exact rank selection inside the boundary bin (ties -> lower index first).
    for (int j = tid; j < nc; j += 256) {
        unsigned kj = fkey(cand_v[j]);
        int      ij = cand_i[j];
        int rank = 0;
        for (int m = 0; m < nc; ++m) {
            unsigned km = fkey(cand_v[m]);
            rank += (km > kj || (km == kj && cand_i[m] < ij)) ? 1 : 0;
        }
        if (rank < need) {
            unsigned p = atomicAdd(&s_nkept, 1u);
            if (p < KEEP_CAP) { kept_v[p] = cand_v[j]; kept_i[p] = ij; }
        }
    }
    __syncthreads();

    // ---------------- Tiny softmax over the <=K kept entries -----------------
    int nk = (int)s_nkept; nk = nk > KEEP_CAP ? KEEP_CAP : nk;
    float maxv = funkey(s_maxkey);
    if (tid < nk) kept_w[tid] = __expf(kept_v[tid] - maxv);   // TEMPERATURE = 1
    __syncthreads();
    if (tid == 0) {
        float Z = 0.f;
        for (int j = 0; j < nk; ++j) Z += kept_w[j];
        s_Z = (Z > 0.f) ? Z : 1.f;
    }
    // Make sure this wave's zero-fill stores completed before any wave scatters.
    asm volatile("s_wait_storecnt 0" ::: "memory");
    __syncthreads();

    float invZ = 1.0f / s_Z;
    if (tid < nk) orow[kept_i[tid]] = kept_w[tid] * invZ;

    // ---------------- Inverse-CDF categorical sample -------------------------
    if (tid == 0) {
        unsigned h = (unsigned)row * 2654435761u;
        h ^= h >> 16; h *= 2246822519u; h ^= h >> 13;
        float u01    = (float)(h >> 8) * (1.0f / 16777216.0f);
        float target = u01 * s_Z;
        float acc = 0.f;
        int   tok = (nk > 0) ? kept_i[nk - 1] : 0;
        for (int j = 0; j < nk; ++j) {
            acc += kept_w[j];
            if (acc >= target) { tok = kept_i[j]; break; }
        }
        tokens[row] = tok;
    }
}

extern "C" void kernel_launch(void* const* d_in, const int* in_sizes, int n_in,
                              void* d_out, int out_size, void* d_ws, size_t ws_size,
                              hipStream_t stream) {
    (void)n_in; (void)out_size; (void)d_ws; (void)ws_size;
    const float* logits = (const float*)d_in[0];
    const int*   pk     = (const int*)d_in[1];
    const int    rows   = in_sizes[0] / VCOLS;   // 8192

    // d_out layout (return order): tokens [rows] then probs [rows * VCOLS].
    int*   tokens = (int*)d_out;
    float* probs  = (float*)d_out + rows;

    topk_softmax_sample_kernel<<<rows, 256, 0, stream>>>(logits, pk, tokens, probs);
}